// Encoder_68152541053661
// MI455X (gfx1250) — compile-verified
//
#include <hip/hip_runtime.h>
#include <hip/hip_bf16.h>
#include <math.h>

// ---------------------------------------------------------------------------
// 2-layer GAT (PyG GATConv, heads=1) for MI455X / gfx1250 (wave32, WMMA).
//
// Pipeline per layer:
//   1) h = x @ W          -> V_WMMA_F32_16X16X4_F32 tiled GEMM (padded cols)
//   2) a_s = h.att_src, a_d = h.att_dst  (wave-per-node dot, shuffle reduce)
//   3) segment max over dst (sign-split int/uint atomic max, init -inf)
//   4) segment sum of exp(e - m[dst])     (native float atomicAdd)
//   5) agg[dst] += alpha * h[src]         (wave-per-edge, lane = feature)
//   6) out = relu(agg + bias), zero-pad for next layer
// Self-loops (src=dst=n) appended implicitly: edge idx >= E maps to node idx-E.
// ---------------------------------------------------------------------------

typedef float v2f __attribute__((ext_vector_type(2)));
typedef float v8f __attribute__((ext_vector_type(8)));

#define NEG_SLOPE 0.2f

// ---------------- weight padding: W[K,F] -> Wp[Kp,Fp] (zero fill) ----------
__global__ void pad_weights_kernel(const float* __restrict__ W, float* __restrict__ Wp,
                                   int K, int F, int Kp, int Fp) {
    int t = blockIdx.x * blockDim.x + threadIdx.x;
    if (t >= Kp * Fp) return;
    int k = t / Fp, f = t % Fp;
    Wp[t] = (k < K && f < F) ? W[k * F + f] : 0.f;
}

// ---------------- WMMA f32 GEMM: C[N,Fp] = A[N,K] * Wp[K,Fp] ---------------
// grid.x = ceil(N/16); blockDim.x = 32 * (Fp/16); wave w owns 16-col tile w.
// A operand (16x4 f32): lane(0..15) v0=K0,v1=K1 ; lane(16..31) v0=K2,v1=K3; M=lane%16
// B operand (4x16 f32): lanes 0-15 hold K0/K1, lanes 16-31 hold K2/K3; N=lane%16
// D (16x16 f32): VGPR i holds M = i + 8*(lane/16), N = lane%16
__global__ void gemm_wmma_kernel(const float* __restrict__ A, const float* __restrict__ Wp,
                                 float* __restrict__ C, int N, int K, int Fp) {
    __shared__ float As[16 * 128];   // K <= 128
    __shared__ float Ws[128 * 64];   // K*Fp <= 8192
    const int row0 = blockIdx.x * 16;
    const int tid  = threadIdx.x;
    const int nthr = blockDim.x;

    for (int t = tid; t < 16 * K; t += nthr) {
        int r = t / K, c = t % K;
        int gr = row0 + r;
        As[t] = (gr < N) ? A[(size_t)gr * K + c] : 0.f;
    }
    for (int t = tid; t < K * Fp; t += nthr) Ws[t] = Wp[t];
    __syncthreads();

    const int lane = tid & 31;
    const int wid  = tid >> 5;
    const int half = lane >> 4;      // K-half selector
    const int m    = lane & 15;      // M (for A) / N (for B,D) index
    const int col0 = wid * 16;

    v8f acc = {};
    for (int kk = 0; kk < K; kk += 4) {
        const int ka = kk + 2 * half;
        v2f a, b;
        a.x = As[m * K + ka];
        a.y = As[m * K + ka + 1];
        b.x = Ws[ka * Fp + col0 + m];
        b.y = Ws[(ka + 1) * Fp + col0 + m];
        acc = __builtin_amdgcn_wmma_f32_16x16x4_f32(
            /*neg_a=*/false, a, /*neg_b=*/false, b,
            /*c_mod=*/(short)0, acc, /*reuse_a=*/false, /*reuse_b=*/false);
    }
#pragma unroll
    for (int i = 0; i < 8; ++i) {
        int gr = row0 + i + 8 * half;
        if (gr < N) C[(size_t)gr * Fp + col0 + m] = acc[i];
    }
}

// ---------------- per-node attention dots (wave per node) ------------------
__global__ void dots_kernel(const float* __restrict__ h,
                            const float* __restrict__ att_s, const float* __restrict__ att_d,
                            float* __restrict__ a_s, float* __restrict__ a_d,
                            int N, int F, int Fp) {
    const int gwave = (blockIdx.x * blockDim.x + threadIdx.x) >> 5;
    const int lane  = threadIdx.x & 31;
    if (gwave >= N) return;
    float ss = 0.f, sd = 0.f;
    for (int f = lane; f < F; f += 32) {
        float v = h[(size_t)gwave * Fp + f];
        ss += v * att_s[f];
        sd += v * att_d[f];
    }
#pragma unroll
    for (int off = 16; off > 0; off >>= 1) {
        ss += __shfl_xor(ss, off, 32);
        sd += __shfl_xor(sd, off, 32);
    }
    if (lane == 0) { a_s[gwave] = ss; a_d[gwave] = sd; }
}

// ---------------- init m=-inf, z=0, agg=0 ----------------------------------
__global__ void init_kernel(float* __restrict__ m, float* __restrict__ z,
                            float* __restrict__ agg, int N, int Fp) {
    int t = blockIdx.x * blockDim.x + threadIdx.x;
    if (t < N) { m[t] = -__builtin_inff(); z[t] = 0.f; }
    if (t < N * Fp) agg[t] = 0.f;
}

__device__ __forceinline__ float leaky(float e) {
    return e > 0.f ? e : NEG_SLOPE * e;
}

// float atomic max via sign-split integer atomics (correct for mixed signs,
// target initialized to -inf).
__device__ __forceinline__ void atomicMaxF(float* addr, float v) {
    if (v >= 0.f) atomicMax((int*)addr, __float_as_int(v));
    else          atomicMin((unsigned int*)addr, __float_as_uint(v));
}

// ---------------- edge pass 1: segment max over dst ------------------------
__global__ void edge_max_kernel(const int* __restrict__ ei,
                                const float* __restrict__ a_s, const float* __restrict__ a_d,
                                float* __restrict__ m, int E, int N) {
    int idx = blockIdx.x * blockDim.x + threadIdx.x;
    if (idx >= E + N) return;
    int s = (idx < E) ? ei[idx]     : (idx - E);
    int d = (idx < E) ? ei[E + idx] : (idx - E);
    atomicMaxF(&m[d], leaky(a_s[s] + a_d[d]));
}

// ---------------- edge pass 2: segment sum of exp(e-m) ---------------------
__global__ void edge_sum_kernel(const int* __restrict__ ei,
                                const float* __restrict__ a_s, const float* __restrict__ a_d,
                                const float* __restrict__ m, float* __restrict__ z,
                                int E, int N) {
    int idx = blockIdx.x * blockDim.x + threadIdx.x;
    if (idx >= E + N) return;
    int s = (idx < E) ? ei[idx]     : (idx - E);
    int d = (idx < E) ? ei[E + idx] : (idx - E);
    atomicAdd(&z[d], __expf(leaky(a_s[s] + a_d[d]) - m[d]));
}

// ---------------- edge pass 3: agg[dst] += alpha * h[src] (wave/edge) ------
__global__ void edge_scatter_kernel(const int* __restrict__ ei,
                                    const float* __restrict__ a_s, const float* __restrict__ a_d,
                                    const float* __restrict__ m, const float* __restrict__ z,
                                    const float* __restrict__ h, float* __restrict__ agg,
                                    int E, int N, int F, int Fp) {
    const int gwave = (blockIdx.x * blockDim.x + threadIdx.x) >> 5;
    const int lane  = threadIdx.x & 31;
    if (gwave >= E + N) return;
    int s = (gwave < E) ? ei[gwave]     : (gwave - E);
    int d = (gwave < E) ? ei[E + gwave] : (gwave - E);
    float alpha = __expf(leaky(a_s[s] + a_d[d]) - m[d]) / z[d];
    for (int f = lane; f < F; f += 32)
        atomicAdd(&agg[(size_t)d * Fp + f], alpha * h[(size_t)s * Fp + f]);
}

// ---------------- finalize: out = relu(agg + bias), zero-pad ---------------
__global__ void finalize_kernel(const float* __restrict__ agg, const float* __restrict__ bias,
                                float* __restrict__ out, int N, int F, int FpIn, int FpOut) {
    int t = blockIdx.x * blockDim.x + threadIdx.x;
    if (t >= N * FpOut) return;
    int n = t / FpOut, f = t % FpOut;
    float v = 0.f;
    if (f < F) {
        v = agg[(size_t)n * FpIn + f] + bias[f];
        v = v > 0.f ? v : 0.f;
    }
    out[t] = v;
}

// ---------------------------------------------------------------------------
extern "C" void kernel_launch(void* const* d_in, const int* in_sizes, int n_in,
                              void* d_out, int out_size, void* d_ws, size_t ws_size,
                              hipStream_t stream) {
    const float* x        = (const float*)d_in[0];
    const int*   ei       = (const int*)d_in[1];
    const float* W1       = (const float*)d_in[2];
    const float* att_s1   = (const float*)d_in[3];
    const float* att_d1   = (const float*)d_in[4];
    const float* bias1    = (const float*)d_in[5];
    const float* W2       = (const float*)d_in[6];
    const float* att_s2   = (const float*)d_in[7];
    const float* att_d2   = (const float*)d_in[8];
    const float* bias2    = (const float*)d_in[9];

    const int H1  = in_sizes[3];            // 50
    const int H2  = in_sizes[7];            // 40
    const int CIN = in_sizes[2] / H1;       // 128
    const int N   = in_sizes[0] / CIN;      // 100000
    const int E   = in_sizes[1] / 2;        // 3200000
    const int Etot = E + N;                 // with self-loops

    const int Fp1 = ((H1 + 15) / 16) * 16;  // 64
    const int Fp2 = ((H2 + 15) / 16) * 16;  // 48

    // ---- workspace carve (floats) ----
    float* ws = (float*)d_ws;
    size_t off = 0;
    float* hbuf  = ws + off; off += (size_t)N * Fp1;   // GEMM output (reused L1/L2)
    float* hin2  = ws + off; off += (size_t)N * Fp1;   // layer-2 input (padded)
    float* agg   = ws + off; off += (size_t)N * Fp1;   // aggregation (reused)
    float* a_s   = ws + off; off += N;
    float* a_d   = ws + off; off += N;
    float* mseg  = ws + off; off += N;
    float* zseg  = ws + off; off += N;
    float* Wp1   = ws + off; off += (size_t)CIN * Fp1;
    float* Wp2   = ws + off; off += (size_t)Fp1 * Fp2;

    const int rowBlocks = (N + 15) / 16;
    const int eThr = (Etot + 255) / 256;          // thread-per-edge grids
    const int eWav = (Etot + 7) / 8;              // wave-per-edge grids (256 thr = 8 waves)
    const int nWav = (N + 7) / 8;                 // wave-per-node grids

    // ---- pad weights ----
    pad_weights_kernel<<<(CIN * Fp1 + 255) / 256, 256, 0, stream>>>(W1, Wp1, CIN, H1, CIN, Fp1);
    pad_weights_kernel<<<(Fp1 * Fp2 + 255) / 256, 256, 0, stream>>>(W2, Wp2, H1, H2, Fp1, Fp2);

    // ================= layer 1 =================
    gemm_wmma_kernel<<<rowBlocks, 32 * (Fp1 / 16), 0, stream>>>(x, Wp1, hbuf, N, CIN, Fp1);
    dots_kernel<<<nWav, 256, 0, stream>>>(hbuf, att_s1, att_d1, a_s, a_d, N, H1, Fp1);
    init_kernel<<<((size_t)N * Fp1 + 255) / 256, 256, 0, stream>>>(mseg, zseg, agg, N, Fp1);
    edge_max_kernel<<<eThr, 256, 0, stream>>>(ei, a_s, a_d, mseg, E, N);
    edge_sum_kernel<<<eThr, 256, 0, stream>>>(ei, a_s, a_d, mseg, zseg, E, N);
    edge_scatter_kernel<<<eWav, 256, 0, stream>>>(ei, a_s, a_d, mseg, zseg, hbuf, agg, E, N, H1, Fp1);
    finalize_kernel<<<((size_t)N * Fp1 + 255) / 256, 256, 0, stream>>>(agg, bias1, hin2, N, H1, Fp1, Fp1);

    // ================= layer 2 =================
    gemm_wmma_kernel<<<rowBlocks, 32 * (Fp2 / 16), 0, stream>>>(hin2, Wp2, hbuf, N, Fp1, Fp2);
    dots_kernel<<<nWav, 256, 0, stream>>>(hbuf, att_s2, att_d2, a_s, a_d, N, H2, Fp2);
    init_kernel<<<((size_t)N * Fp2 + 255) / 256, 256, 0, stream>>>(mseg, zseg, agg, N, Fp2);
    edge_max_kernel<<<eThr, 256, 0, stream>>>(ei, a_s, a_d, mseg, E, N);
    edge_sum_kernel<<<eThr, 256, 0, stream>>>(ei, a_s, a_d, mseg, zseg, E, N);
    edge_scatter_kernel<<<eWav, 256, 0, stream>>>(ei, a_s, a_d, mseg, zseg, hbuf, agg, E, N, H2, Fp2);
    finalize_kernel<<<((size_t)N * H2 + 255) / 256, 256, 0, stream>>>(agg, bias2, (float*)d_out, N, H2, Fp2, H2);

    // ---- tuple output: append edge_index bit-exact after h [N, H2] ----
    hipMemcpyAsync((char*)d_out + (size_t)N * H2 * sizeof(float), d_in[1],
                   (size_t)in_sizes[1] * sizeof(int), hipMemcpyDeviceToDevice, stream);
}